// TripletLoss_29841432773241
// MI455X (gfx1250) — compile-verified
//
#include <hip/hip_runtime.h>
#include <hip/hip_bf16.h>

typedef __attribute__((ext_vector_type(2))) float v2f;
typedef __attribute__((ext_vector_type(8))) float v8f;

#define B_N    4096
#define D_K    512
#define MARGIN 1.0f
#define BIGF   1.0e9f
#define TILE   64          // block output tile (64x64)
#define KC     32          // K chunk staged through LDS
#define LDSTR  (KC + 4)    // padded LDS row stride (floats) to avoid bank conflicts

// ---------------------------------------------------------------------------
// Kernel 1: per-row squared norms + init of the hardest-pos/neg accumulators.
// One wave32 per row; 8 rows per 256-thread block.
// hp/hn use the monotonic IEEE bit trick (all candidates >= 0):
//   max over floats == max over u32 bits;  min likewise.
// ---------------------------------------------------------------------------
__global__ __launch_bounds__(256) void sq_init_kernel(
    const float* __restrict__ E, float* __restrict__ sq,
    unsigned* __restrict__ hp, unsigned* __restrict__ hn) {
  int row  = blockIdx.x * 8 + (threadIdx.x >> 5);
  int lane = threadIdx.x & 31;
  const float* e = E + (size_t)row * D_K;
  float s = 0.0f;
  #pragma unroll 4
  for (int k = lane; k < D_K; k += 32) { float v = e[k]; s += v * v; }
  #pragma unroll
  for (int m = 16; m >= 1; m >>= 1) s += __shfl_xor(s, m, 32);
  if (lane == 0) {
    sq[row] = s;
    hp[row] = 0u;           // identity for max of nonnegative distances
    hn[row] = 0x7F7FFFFFu;  // FLT_MAX bits: identity for min
  }
}

// ---------------------------------------------------------------------------
// Fused epilogue for one 16x16 accumulator tile:
//   d = sqrt(max(sq_i - 2g + sq_j, 0) + 1e-16)
//   pos candidate: same-class & off-diagonal -> d, else 0  (row max)
//   neg candidate: d + (same-class ? BIG : 0)              (row min)
// C/D layout (ISA 7.12.2): VGPR v, lanes 0-15 -> M=v, lanes 16-31 -> M=v+8,
// N = lane&15. So each output row lives in 16 lanes of one half-wave:
// reduce with xor masks 1,2,4,8 (stays within the half), then one atomic.
// ---------------------------------------------------------------------------
__device__ __forceinline__ void epilogue_tile(
    const v8f& acc, int ibase, int jbase, int m, int h,
    const float* __restrict__ sq, const int* __restrict__ labels,
    float sqj, int lj, unsigned* __restrict__ hp, unsigned* __restrict__ hn) {
  int j = jbase + m;
  #pragma unroll
  for (int v = 0; v < 8; ++v) {
    int   i  = ibase + v + 8 * h;
    float g  = acc[v];
    float d2 = sq[i] - 2.0f * g + sqj;
    float d  = sqrtf(fmaxf(d2, 0.0f) + 1e-16f);
    int   li = labels[i];
    bool  same = (li == lj);
    float pos = (same && (i != j)) ? d : 0.0f;
    float neg = d + (same ? BIGF : 0.0f);
    #pragma unroll
    for (int msk = 8; msk >= 1; msk >>= 1) {
      pos = fmaxf(pos, __shfl_xor(pos, msk, 32));
      neg = fminf(neg, __shfl_xor(neg, msk, 32));
    }
    if (m == 0) {
      atomicMax(&hp[i], __float_as_uint(pos));
      atomicMin(&hn[i], __float_as_uint(neg));
    }
  }
}

// ---------------------------------------------------------------------------
// Kernel 2: tiled G = E*E^T via V_WMMA_F32_16X16X4_F32, fused distance +
// hard-mining epilogue. Block = 256 threads = 8 waves; block tile 64x64.
// Wave w -> (wi = w>>2 in 0..1, wj = w&3 in 0..3); each wave computes two
// stacked 16x16 subtiles (rows wi*32 .. wi*32+31, cols wj*16 .. wj*16+15),
// reusing its B fragment for both.
// ---------------------------------------------------------------------------
__global__ __launch_bounds__(256) void dist_tile_kernel(
    const float* __restrict__ E, const int* __restrict__ labels,
    const float* __restrict__ sq,
    unsigned* __restrict__ hp, unsigned* __restrict__ hn) {
  __shared__ __align__(16) float lA[TILE * LDSTR];
  __shared__ __align__(16) float lB[TILE * LDSTR];

  const int ib = blockIdx.y * TILE;   // output row block (i)
  const int jb = blockIdx.x * TILE;   // output col block (j)
  const int t    = threadIdx.x;
  const int w    = t >> 5;
  const int lane = t & 31;
  const int wi   = w >> 2;            // 0..1
  const int wj   = w & 3;             // 0..3
  const int m    = lane & 15;         // row-in-tile for A frag / N for B frag
  const int h    = lane >> 4;         // lane half selects K sub-pair

  // cooperative loader mapping: 32 rows x 32 cols (float4) per pass, 2 passes
  const int lrow = t >> 3;            // 0..31
  const int lcol = (t & 7) * 4;       // 0,4,...,28

  v8f acc0 = {};  // rows ib + wi*32 + 0..15
  v8f acc1 = {};  // rows ib + wi*32 + 16..31

  for (int k0 = 0; k0 < D_K; k0 += KC) {
    __syncthreads();
    #pragma unroll
    for (int half = 0; half < 2; ++half) {
      int r = lrow + 32 * half;
      float4 a4 = *(const float4*)(E + (size_t)(ib + r) * D_K + k0 + lcol);
      float4 b4 = *(const float4*)(E + (size_t)(jb + r) * D_K + k0 + lcol);
      *(float4*)&lA[r * LDSTR + lcol] = a4;
      *(float4*)&lB[r * LDSTR + lcol] = b4;
      if (k0 + KC < D_K) {  // prefetch next K slab -> global_prefetch_b8
        __builtin_prefetch(E + (size_t)(ib + r) * D_K + k0 + KC + lcol, 0, 0);
        __builtin_prefetch(E + (size_t)(jb + r) * D_K + k0 + KC + lcol, 0, 0);
      }
    }
    __syncthreads();

    // A frag (16x4 f32, ISA layout): lanes 0-15 hold K={0,1}, 16-31 K={2,3}
    #pragma unroll
    for (int ks = 0; ks < KC; ks += 4) {
      int kk = ks + h * 2;
      v2f a0, a1, b;
      a0.x = lA[(wi * 32 +      m) * LDSTR + kk];
      a0.y = lA[(wi * 32 +      m) * LDSTR + kk + 1];
      a1.x = lA[(wi * 32 + 16 + m) * LDSTR + kk];
      a1.y = lA[(wi * 32 + 16 + m) * LDSTR + kk + 1];
      // B frag (4x16): B[k][n] = E[jb + n][k], n = m
      b.x  = lB[(wj * 16 + m) * LDSTR + kk];
      b.y  = lB[(wj * 16 + m) * LDSTR + kk + 1];
      acc0 = __builtin_amdgcn_wmma_f32_16x16x4_f32(
                 false, a0, false, b, (short)0, acc0, false, false);
      acc1 = __builtin_amdgcn_wmma_f32_16x16x4_f32(
                 false, a1, false, b, (short)0, acc1, false, false);
    }
  }

  // fused distance + hard-mining epilogue
  const int jcol = jb + wj * 16 + m;
  const float sqj = sq[jcol];
  const int   lj  = labels[jcol];
  epilogue_tile(acc0, ib + wi * 32,      jb + wj * 16, m, h, sq, labels, sqj, lj, hp, hn);
  epilogue_tile(acc1, ib + wi * 32 + 16, jb + wj * 16, m, h, sq, labels, sqj, lj, hp, hn);
}

// ---------------------------------------------------------------------------
// Kernel 3: loss_i = relu(hp_i - hn_i + margin); output mean over B rows.
// ---------------------------------------------------------------------------
__global__ __launch_bounds__(1024) void finalize_kernel(
    const unsigned* __restrict__ hp, const unsigned* __restrict__ hn,
    float* __restrict__ out) {
  __shared__ float red[32];
  int t = threadIdx.x;
  float s = 0.0f;
  for (int r = t; r < B_N; r += 1024) {
    float l = __uint_as_float(hp[r]) - __uint_as_float(hn[r]) + MARGIN;
    s += (l > 0.0f) ? l : 0.0f;
  }
  #pragma unroll
  for (int msk = 16; msk >= 1; msk >>= 1) s += __shfl_xor(s, msk, 32);
  if ((t & 31) == 0) red[t >> 5] = s;
  __syncthreads();
  if (t < 32) {
    float v = red[t];
    #pragma unroll
    for (int msk = 16; msk >= 1; msk >>= 1) v += __shfl_xor(v, msk, 32);
    if (t == 0) *out = v / (float)B_N;
  }
}

// ---------------------------------------------------------------------------
extern "C" void kernel_launch(void* const* d_in, const int* in_sizes, int n_in,
                              void* d_out, int out_size, void* d_ws, size_t ws_size,
                              hipStream_t stream) {
  (void)in_sizes; (void)n_in; (void)out_size; (void)ws_size;
  const float* E      = (const float*)d_in[0];
  const int*   labels = (const int*)d_in[1];

  // workspace: [sq: 4096 f32][hp: 4096 u32][hn: 4096 u32] = 48 KB
  float*    sq = (float*)d_ws;
  unsigned* hp = (unsigned*)d_ws + B_N;
  unsigned* hn = (unsigned*)d_ws + 2 * B_N;

  sq_init_kernel<<<B_N / 8, 256, 0, stream>>>(E, sq, hp, hn);

  dim3 grid(B_N / TILE, B_N / TILE);
  dist_tile_kernel<<<grid, 256, 0, stream>>>(E, labels, sq, hp, hn);

  finalize_kernel<<<1, 1024, 0, stream>>>(hp, hn, (float*)d_out);
}